// WideLayer_37014028157486
// MI455X (gfx1250) — compile-verified
//
#include <hip/hip_runtime.h>
#include <stdint.h>

#define N_FEAT  512
#define N_PAIRS (N_FEAT * (N_FEAT - 1) / 2)   // 130816
#define NTILE   (N_FEAT / 16)                 // 32 tiles per side
#define N_TRI_TILES (NTILE * (NTILE + 1) / 2) // 528 upper-tri tiles (incl. diagonal)
#define WAVES_PER_BLOCK 8
#define TILE_GROUPS (N_TRI_TILES / WAVES_PER_BLOCK) // 66
#define BATCH_PER_BLOCK 8

typedef __attribute__((ext_vector_type(2))) float v2f;
typedef __attribute__((ext_vector_type(8))) float v8f;
typedef __attribute__((address_space(1))) float gfloat;   // explicit global AS

// Issue a 2 KB async global->LDS copy of one x row (threads 0..127, 16B/lane).
__device__ __forceinline__ void async_stage_row(const float* __restrict__ src,
                                                float* dst_lds, int tid)
{
    if (tid < 128) {
        unsigned lds_off = (unsigned)(uintptr_t)
            (__attribute__((address_space(3))) float*)(dst_lds + tid * 4);
        unsigned long long gaddr = (unsigned long long)(uintptr_t)(src + tid * 4);
        asm volatile("global_load_async_to_lds_b128 %0, %1, off"
                     :: "v"(lds_off), "v"(gaddr) : "memory");
    }
}

__device__ __forceinline__ void async_wait(int tid)
{
    if (tid < 128) {
        asm volatile("s_wait_asynccnt 0" ::: "memory");
    }
}

__global__ __launch_bounds__(256)
void widelayer_pair_kernel(const float* __restrict__ x,
                           const float* __restrict__ weight,
                           float* __restrict__ out,
                           int batch_total)
{
    __shared__ float xs[2][N_FEAT];               // double-buffered x row, 2x2 KB
    const int tid  = threadIdx.x;
    const int lane = tid & 31;
    const int wave = tid >> 5;

    // ---- decode this wave's 16x16 tile (ti <= tj upper triangle) ----
    int t = blockIdx.x * WAVES_PER_BLOCK + wave;  // 0..527, uniform per wave
    int ti = 0, rem = t;
    while (rem >= NTILE - ti) { rem -= NTILE - ti; ++ti; }
    const int tj = ti + rem;
    const int i0 = ti * 16, j0 = tj * 16;
    const bool full_tile = (ti != tj);            // wave-uniform: all 256 pairs valid

    const int b0 = blockIdx.y * BATCH_PER_BLOCK;
    const int nb = min(BATCH_PER_BLOCK, batch_total - b0);

    // ---- batch-invariant: weights + running output pointers for 8 outputs ----
    const int n    = lane & 15;          // N within tile
    const int mb   = (lane >> 4) << 3;   // M base: 0 or 8 (D-matrix VGPR layout)
    const bool hi  = (lane >= 16);
    float   wv[8];
    gfloat* op[8];                       // running GLOBAL out pointers
    bool    ok[8];
    #pragma unroll
    for (int r = 0; r < 8; ++r) {
        const int i = i0 + mb + r;
        const int j = j0 + n;
        ok[r] = (j > i);
        const int p = ok[r] ? (((i * (2 * N_FEAT - 1 - i)) >> 1) + (j - i - 1)) : 0;
        wv[r] = ok[r] ? weight[p] : 0.0f;    // 0.5 MB table, L2-resident
        op[r] = (gfloat*)(out + (size_t)b0 * N_PAIRS + p);
    }

    // ---- prologue: stage first x row ----
    async_stage_row(x + (size_t)b0 * N_FEAT, xs[0], tid);
    async_wait(tid);
    __syncthreads();

    for (int bi = 0; bi < nb; ++bi) {
        const float* cur = xs[bi & 1];

        // ---- prefetch next x row into the other buffer (pipelined) ----
        if (bi + 1 < nb) {
            async_stage_row(x + (size_t)(b0 + bi + 1) * N_FEAT, xs[(bi + 1) & 1], tid);
        }

        // ---- rank-1 outer product via WMMA: D[m][n] = x[i0+m] * x[j0+n] ----
        // Unconditional LDS loads from clamped index; zero upper lanes via select.
        const float ar = cur[i0 + n];    // i0 + (lane & 15), always in-bounds
        const float br = cur[j0 + n];
        v2f a, bmat;
        a.x    = hi ? 0.0f : ar;
        a.y    = 0.0f;
        bmat.x = hi ? 0.0f : br;
        bmat.y = 0.0f;
        v8f c = {};
        v8f d = __builtin_amdgcn_wmma_f32_16x16x4_f32(
            /*neg_a=*/false, a, /*neg_b=*/false, bmat,
            /*c_mod=*/(short)0, c, /*reuse_a=*/false, /*reuse_b=*/false);

        // ---- scale by weight and stream out non-temporally (global AS) ----
        if (full_tile) {
            // 496/528 tiles: unconditional stores, no exec-mask churn
            #pragma unroll
            for (int r = 0; r < 8; ++r) {
                __builtin_nontemporal_store(d[r] * wv[r], op[r]);
            }
        } else {
            // 32 diagonal tiles: keep j > i predicate
            #pragma unroll
            for (int r = 0; r < 8; ++r) {
                if (ok[r]) {
                    __builtin_nontemporal_store(d[r] * wv[r], op[r]);
                }
            }
        }

        // ---- advance output pointers to next batch row ----
        #pragma unroll
        for (int r = 0; r < 8; ++r) {
            op[r] += N_PAIRS;
        }

        // ---- retire prefetch before it is consumed next iteration ----
        if (bi + 1 < nb) {
            async_wait(tid);
            __syncthreads();
        }
    }
}

extern "C" void kernel_launch(void* const* d_in, const int* in_sizes, int n_in,
                              void* d_out, int out_size, void* d_ws, size_t ws_size,
                              hipStream_t stream)
{
    const float* x      = (const float*)d_in[0];
    const float* weight = (const float*)d_in[1];
    float* out          = (float*)d_out;

    const int B  = in_sizes[0] / N_FEAT;  // 1024
    const int by = (B + BATCH_PER_BLOCK - 1) / BATCH_PER_BLOCK;
    dim3 grid(TILE_GROUPS, by);           // 66 x 128 blocks
    widelayer_pair_kernel<<<grid, 256, 0, stream>>>(x, weight, out, B);
}